// Hitsbe_37546604102113
// MI455X (gfx1250) — compile-verified
//
#include <hip/hip_runtime.h>
#include <hip/hip_bf16.h>

typedef float v2f __attribute__((ext_vector_type(2)));
typedef float v4f __attribute__((ext_vector_type(4)));
typedef float v8f __attribute__((ext_vector_type(8)));

#define NB     256
#define NSEQ   128
#define NV     1024
#define NCELL  8
#define ND     768
#define NROWS  (NB * NSEQ)          // 32768
#define THRESH 0.95f

// ---------------------------------------------------------------------------
// Kernel 1: per-batch prep. 1 workgroup (128 threads) per batch row.
// ---------------------------------------------------------------------------
__global__ __launch_bounds__(128) void prep_batch(const float* __restrict__ X,
                                                  float* __restrict__ seg_c,
                                                  float* __restrict__ seg_rstd,
                                                  float* __restrict__ hc) {
    __shared__ float a[1024];
    __shared__ float w0[512];
    __shared__ float w1[256];
    __shared__ float dstore[255];   // cD_3(128) cD_4(64) ... cD_10(1)

    const int b = blockIdx.x;
    const int t = threadIdx.x;
    const float INV = 0.70710678118654752440f;   // 1/sqrt(2)

    for (int i = t; i < 1024; i += 128) a[i] = X[b * 1024 + i];
    __syncthreads();

    // --- segment stats (thread t == segment s) ---
    {
        const float* xs = &a[t * NCELL];
        float sum = 0.f;
        #pragma unroll
        for (int i = 0; i < NCELL; ++i) sum += xs[i];
        const float mean = sum * 0.125f;
        float var = 0.f;
        float c[NCELL];
        #pragma unroll
        for (int i = 0; i < NCELL; ++i) { c[i] = xs[i] - mean; var += c[i] * c[i]; }
        const int row = b * NSEQ + t;
        #pragma unroll
        for (int i = 0; i < NCELL; ++i) seg_c[(size_t)row * NCELL + i] = c[i];
        const float sd = sqrtf(var * 0.125f);
        seg_rstd[row] = (sd == 0.0f) ? 0.0f : (1.0f / sd);   // guarded reciprocal
    }

    // --- Haar cascade ---
    for (int j = t; j < 512; j += 128) w0[j] = (a[2 * j] + a[2 * j + 1]) * INV;
    __syncthreads();
    for (int j = t; j < 256; j += 128) w1[j] = (w0[2 * j] + w0[2 * j + 1]) * INV;
    __syncthreads();

    {
        float* src = w1;
        float* dst = w0;
        int len = 128, off = 0;
        for (int lev = 3; lev <= 10; ++lev) {
            if (t < len) {
                const float e = src[2 * t], o = src[2 * t + 1];
                dstore[off + t] = (e - o) * INV;   // detail
                dst[t]          = (e + o) * INV;   // approx
            }
            __syncthreads();
            float* tmp = src; src = dst; dst = tmp;
            off += len; len >>= 1;
        }
    }

    // --- hc[b, s, l] = cD_{10-l}[ s >> (7-l) ] ---
    {
        const int offs[8] = {0, 128, 192, 224, 240, 248, 252, 254};
        const int row = b * NSEQ + t;
        #pragma unroll
        for (int l = 0; l < 8; ++l) {
            hc[(size_t)row * 8 + l] = dstore[offs[7 - l] + (t >> (7 - l))];
        }
    }
}

// ---------------------------------------------------------------------------
// Kernel 2: vocab prep. Centered words + guarded (0.125 / std).
// ---------------------------------------------------------------------------
__global__ __launch_bounds__(128) void prep_vocab(const float* __restrict__ vocab,
                                                  float* __restrict__ w_c,
                                                  float* __restrict__ w_rs) {
    const int v = blockIdx.x * 128 + threadIdx.x;
    if (v >= NV) return;
    const float* xs = vocab + (size_t)v * NCELL;
    float sum = 0.f;
    #pragma unroll
    for (int i = 0; i < NCELL; ++i) sum += xs[i];
    const float mean = sum * 0.125f;
    float var = 0.f;
    float c[NCELL];
    #pragma unroll
    for (int i = 0; i < NCELL; ++i) { c[i] = xs[i] - mean; var += c[i] * c[i]; }
    #pragma unroll
    for (int i = 0; i < NCELL; ++i) w_c[(size_t)v * NCELL + i] = c[i];
    const float sd = sqrtf(var * 0.125f);
    w_rs[v] = (sd == 0.0f) ? 0.0f : (0.125f / sd);
}

// ---------------------------------------------------------------------------
// Kernel 3: correlation (WMMA f32 16x16x4, K=8) + exact sequential selection.
// Each wave owns 16 rows. Scan reads its LDS row with b128 loads into regs,
// then a log-depth max test skips the serial state machine when no take is
// possible for this 16-value chunk (|c|<=0.95 everywhere, or max|c|<=best).
// ---------------------------------------------------------------------------
__global__ __launch_bounds__(128) void corr_select(const float* __restrict__ seg_c,
                                                   const float* __restrict__ seg_rstd,
                                                   const float* __restrict__ w_c,
                                                   const float* __restrict__ w_rs,
                                                   int* __restrict__ idx_out) {
    __shared__ float scanbuf[4][16 * 16];
    __shared__ float srs_lds[4][16];

    const int wave = threadIdx.x >> 5;
    const int lane = threadIdx.x & 31;
    const int m    = lane & 15;     // A-row / B-col within tile
    const int kh   = lane >> 4;     // which K pair this lane holds
    const int R0   = (blockIdx.x * 4 + wave) * 16;

    if (lane < 16) srs_lds[wave][lane] = seg_rstd[R0 + lane];
    __syncthreads();

    v2f A1, A2;
    {
        const float* srow = seg_c + (size_t)(R0 + m) * NCELL;
        A1[0] = srow[2 * kh];     A1[1] = srow[2 * kh + 1];
        A2[0] = srow[4 + 2 * kh]; A2[1] = srow[5 + 2 * kh];
    }
    float srs_r[8];
    #pragma unroll
    for (int r = 0; r < 8; ++r) srs_r[r] = srs_lds[wave][r + 8 * kh];

    float best = -__builtin_inff();
    int   bidx = -1;

    for (int tI = 0; tI < 64; ++tI) {            // vocab tiles, increasing v
        const int v = tI * 16 + m;
        const float* wrow = w_c + (size_t)v * NCELL;
        v2f B1, B2;
        B1[0] = wrow[2 * kh];     B1[1] = wrow[2 * kh + 1];
        B2[0] = wrow[4 + 2 * kh]; B2[1] = wrow[5 + 2 * kh];
        const float wrs = w_rs[v];               // includes 1/CELL

        v8f acc = {};
        acc = __builtin_amdgcn_wmma_f32_16x16x4_f32(false, A1, false, B1,
                                                    (short)0, acc, false, false);
        acc = __builtin_amdgcn_wmma_f32_16x16x4_f32(false, A2, false, B2,
                                                    (short)0, acc, false, false);

        #pragma unroll
        for (int r = 0; r < 8; ++r) {
            scanbuf[wave][(r + 8 * kh) * 16 + m] = acc[r] * wrs * srs_r[r];
        }
        asm volatile("s_wait_dscnt 0x0" ::: "memory");

        if (lane < 16) {
            // batch-load the row chunk (64B contiguous -> ds_load_b128 x4)
            const v4f* rowp = (const v4f*)&scanbuf[wave][lane * 16];
            v4f q0 = rowp[0], q1 = rowp[1], q2 = rowp[2], q3 = rowp[3];
            float c[16];
            #pragma unroll
            for (int j = 0; j < 4; ++j) {
                c[j] = q0[j]; c[4 + j] = q1[j]; c[8 + j] = q2[j]; c[12 + j] = q3[j];
            }
            // log-depth max|c| to decide whether any take is possible
            float mx[8];
            #pragma unroll
            for (int j = 0; j < 8; ++j) mx[j] = fmaxf(fabsf(c[j]), fabsf(c[8 + j]));
            #pragma unroll
            for (int j = 0; j < 4; ++j) mx[j] = fmaxf(mx[j], mx[4 + j]);
            mx[0] = fmaxf(mx[0], mx[2]);
            mx[1] = fmaxf(mx[1], mx[3]);
            const float mxall = fmaxf(mx[0], mx[1]);

            if (mxall > THRESH && mxall > best) {
                // rare path: exact sequential state machine over this chunk
                #pragma unroll
                for (int j = 0; j < 16; ++j) {
                    const float aabs = fabsf(c[j]);
                    if (aabs > THRESH && best < aabs) { best = c[j]; bidx = tI * 16 + j; }
                }
            }
        }
    }
    if (lane < 16) idx_out[R0 + lane] = bidx;
}

// ---------------------------------------------------------------------------
// Kernel 4: out[row, d] = (hc*mask) @ haar_emb + pos_emb[s] + mask*word_emb[idx]
// Each wave covers 16 rows x 96 cols (6 col tiles) to amortize A/idx loads.
// ---------------------------------------------------------------------------
__global__ __launch_bounds__(128) void output_fused(const float* __restrict__ hc,
                                                    const int* __restrict__ idx,
                                                    const float* __restrict__ haar_emb,
                                                    const float* __restrict__ pos_emb,
                                                    const float* __restrict__ word_emb,
                                                    float* __restrict__ out) {
    const int wave = threadIdx.x >> 5;
    const int lane = threadIdx.x & 31;
    const int unit = blockIdx.x * 4 + wave;    // 0..16383
    const int rb = unit >> 3;                  // 16-row block (0..2047)
    const int cg = unit & 7;                   // 96-col group (0..7)
    const int m  = lane & 15;
    const int kh = lane >> 4;
    const int arow = rb * 16 + m;              // A-matrix row for this lane

    // A: masked haar coefficients (loaded once, reused for 6 tiles)
    const float maskA = (idx[arow] >= 0) ? 1.0f : 0.0f;
    const float* hrow = hc + (size_t)arow * 8;
    const v2f A1 = { hrow[2 * kh] * maskA,     hrow[2 * kh + 1] * maskA };
    const v2f A2 = { hrow[4 + 2 * kh] * maskA, hrow[5 + 2 * kh] * maskA };

    // per-output-row word indices (reused for 6 tiles)
    int wi_r[8];
    #pragma unroll
    for (int r = 0; r < 8; ++r) wi_r[r] = idx[rb * 16 + r + 8 * kh];

    #pragma unroll 1
    for (int ct2 = 0; ct2 < 6; ++ct2) {
        const int col = (cg * 6 + ct2) * 16 + m;

        // B: haar_emb is already [K=8][N=768]
        v2f B1 = { haar_emb[(2 * kh) * ND + col],     haar_emb[(2 * kh + 1) * ND + col] };
        v2f B2 = { haar_emb[(4 + 2 * kh) * ND + col], haar_emb[(5 + 2 * kh) * ND + col] };

        v8f acc = {};
        acc = __builtin_amdgcn_wmma_f32_16x16x4_f32(false, A1, false, B1,
                                                    (short)0, acc, false, false);
        acc = __builtin_amdgcn_wmma_f32_16x16x4_f32(false, A2, false, B2,
                                                    (short)0, acc, false, false);

        #pragma unroll
        for (int r = 0; r < 8; ++r) {
            const int grow = rb * 16 + r + 8 * kh;         // global output row
            const int s    = grow & (NSEQ - 1);
            const float we = (wi_r[r] >= 0)
                                 ? word_emb[(size_t)wi_r[r] * ND + col] : 0.0f;
            out[(size_t)grow * ND + col] = acc[r] + pos_emb[s * ND + col] + we;
        }
    }
}

// ---------------------------------------------------------------------------
extern "C" void kernel_launch(void* const* d_in, const int* in_sizes, int n_in,
                              void* d_out, int out_size, void* d_ws, size_t ws_size,
                              hipStream_t stream) {
    const float* X        = (const float*)d_in[0];   // [256,1024]
    const float* vocab    = (const float*)d_in[1];   // [1024,8]
    const float* word_emb = (const float*)d_in[2];   // [1024,768]
    const float* haar_emb = (const float*)d_in[3];   // [8,768]
    const float* pos_emb  = (const float*)d_in[4];   // [128,768]
    float* out = (float*)d_out;                      // [256,128,768]

    float* ws       = (float*)d_ws;
    float* seg_c    = ws;                       // 32768*8 = 262144
    float* seg_rstd = ws + 262144;              // 32768
    float* hc       = ws + 294912;              // 262144
    float* w_c      = ws + 557056;              // 8192
    float* w_rs     = ws + 565248;              // 1024
    int*   idxb     = (int*)(ws + 566272);      // 32768 ints  (~2.4 MB total)

    prep_batch<<<NB, 128, 0, stream>>>(X, seg_c, seg_rstd, hc);
    prep_vocab<<<NV / 128, 128, 0, stream>>>(vocab, w_c, w_rs);
    // 32768 rows / 16 per wave = 2048 waves; 4 waves per WG -> 512 WGs
    corr_select<<<512, 128, 0, stream>>>(seg_c, seg_rstd, w_c, w_rs, idxb);
    // 2048 row-blocks * 8 col-groups = 16384 wave-units; 4 per WG -> 4096 WGs
    output_fused<<<4096, 128, 0, stream>>>(hc, idxb, haar_emb, pos_emb, word_emb, out);
}